// JanossyReadout_30502857736352
// MI455X (gfx1250) — compile-verified
//
#include <hip/hip_runtime.h>
#include <hip/hip_bf16.h>

typedef __attribute__((ext_vector_type(2))) float v2f;
typedef __attribute__((ext_vector_type(8))) float v8f;

#define N_FRAG  262144
#define MBLOCK  32          // fragments per workgroup
#define KDIM    384
#define DMID    256
#define HSTR    396         // padded LDS stride for gathered features (conflict-free, 16B aligned)
#define ZSTR    268         // padded LDS stride for Z1 / pooled (conflict-free, 16B aligned)

__device__ __forceinline__ v8f wmma_f32(v2f a, v2f b, v8f c) {
    return __builtin_amdgcn_wmma_f32_16x16x4_f32(
        /*neg_a=*/false, a, /*neg_b=*/false, b,
        /*c_mod=*/(short)0, c, /*reuse_a=*/false, /*reuse_b=*/false);
}

__global__ __launch_bounds__(128, 1)
void janossy_readout_kernel(const float* __restrict__ x_atoms,
                            const int*   __restrict__ frag_idx,
                            const float* __restrict__ W1,
                            const float* __restrict__ b1,
                            const float* __restrict__ W2,
                            const float* __restrict__ b2,
                            const float* __restrict__ Wout,
                            const float* __restrict__ bout,
                            float* __restrict__ out)
{
    __shared__ float sh_h  [MBLOCK * HSTR];   // gathered fwd features [32][384+pad]
    __shared__ float sh_z1f[MBLOCK * ZSTR];   // relu(fwd@W1+b1)
    __shared__ float sh_z1r[MBLOCK * ZSTR];   // relu(rev@W1+b1)
    __shared__ float sh_p  [MBLOCK * ZSTR];   // pooled layer-2 output
    __shared__ int   sh_idx[MBLOCK * 3];

    const int tid   = threadIdx.x;
    const int frag0 = blockIdx.x * MBLOCK;

    // ---- stage 0: fragment atom indices ----
    if (tid < MBLOCK * 3) sh_idx[tid] = frag_idx[frag0 * 3 + tid];
    __syncthreads();

    // ---- stage 1: gather 96 atom rows (32 frags x 3 slots x 128 f32) into LDS ----
    for (int i = tid; i < MBLOCK * 3 * 32; i += 128) {
        int row = i >> 5;           // 0..95
        int c4  = i & 31;           // float4 index within 128-wide row
        int m   = row / 3;
        int k   = row - m * 3;
        const float4* src = (const float4*)(x_atoms + (size_t)sh_idx[row] * 128);
        float4 v = src[c4];
        *(float4*)&sh_h[m * HSTR + k * 128 + c4 * 4] = v;
    }
    __syncthreads();

    const int lane  = tid & 31;
    const int wave  = tid >> 5;
    const int mrow  = lane & 15;    // A-matrix: M = lane%16
    const int khalf = lane >> 4;    // A/B-matrix: K offset 2*(lane/16)
    const int ncol  = lane & 15;    // B/C/D: N = lane%16
    const int n0    = wave * 64;    // this wave's 64-wide N chunk

    v8f accF[2][4], accR[2][4];
    #pragma unroll
    for (int mt = 0; mt < 2; ++mt)
        #pragma unroll
        for (int t = 0; t < 4; ++t) {
            accF[mt][t] = (v8f){0.f,0.f,0.f,0.f,0.f,0.f,0.f,0.f};
            accR[mt][t] = (v8f){0.f,0.f,0.f,0.f,0.f,0.f,0.f,0.f};
        }

    // ---- stage 2: layer 1, K = 384 in steps of 4 ----
    // rev branch reuses the same W1 B-tile; its A-tile comes from the
    // block-reversed K offset:  kr = (2 - k/128)*128 + k%128
    for (int kb = 0; kb < KDIM / 4; ++kb) {
        const int kf = kb * 4;
        const int kr = (2 - (kb >> 5)) * 128 + (kf & 127);
        const int ka  = kf + 2 * khalf;   // this lane's K pair (fwd)
        const int kra = kr + 2 * khalf;   // this lane's K pair (rev)

        v2f aF0 = *(const v2f*)&sh_h[ mrow       * HSTR + ka ];
        v2f aF1 = *(const v2f*)&sh_h[(16 + mrow) * HSTR + ka ];
        v2f aR0 = *(const v2f*)&sh_h[ mrow       * HSTR + kra];
        v2f aR1 = *(const v2f*)&sh_h[(16 + mrow) * HSTR + kra];

        #pragma unroll
        for (int t = 0; t < 4; ++t) {
            const int n = n0 + t * 16 + ncol;
            v2f b;
            b.x = W1[(ka + 0) * DMID + n];
            b.y = W1[(ka + 1) * DMID + n];
            accF[0][t] = wmma_f32(aF0, b, accF[0][t]);
            accF[1][t] = wmma_f32(aF1, b, accF[1][t]);
            accR[0][t] = wmma_f32(aR0, b, accR[0][t]);
            accR[1][t] = wmma_f32(aR1, b, accR[1][t]);
        }
    }

    // ---- bias + relu, write Z1 to LDS (C/D layout: M = i + 8*khalf, N = ncol) ----
    #pragma unroll
    for (int mt = 0; mt < 2; ++mt)
        #pragma unroll
        for (int t = 0; t < 4; ++t) {
            const int n = n0 + t * 16 + ncol;
            const float bias = b1[n];
            #pragma unroll
            for (int i = 0; i < 8; ++i) {
                const int m = mt * 16 + i + 8 * khalf;
                float vf = accF[mt][t][i] + bias; vf = vf > 0.f ? vf : 0.f;
                float vr = accR[mt][t][i] + bias; vr = vr > 0.f ? vr : 0.f;
                sh_z1f[m * ZSTR + n] = vf;
                sh_z1r[m * ZSTR + n] = vr;
            }
        }
    __syncthreads();

    #pragma unroll
    for (int mt = 0; mt < 2; ++mt)
        #pragma unroll
        for (int t = 0; t < 4; ++t) {
            accF[mt][t] = (v8f){0.f,0.f,0.f,0.f,0.f,0.f,0.f,0.f};
            accR[mt][t] = (v8f){0.f,0.f,0.f,0.f,0.f,0.f,0.f,0.f};
        }

    // ---- stage 3: layer 2, K = 256 in steps of 4 (shared B tile for fwd/rev) ----
    for (int kb = 0; kb < DMID / 4; ++kb) {
        const int ka = kb * 4 + 2 * khalf;

        v2f aF0 = *(const v2f*)&sh_z1f[ mrow       * ZSTR + ka];
        v2f aF1 = *(const v2f*)&sh_z1f[(16 + mrow) * ZSTR + ka];
        v2f aR0 = *(const v2f*)&sh_z1r[ mrow       * ZSTR + ka];
        v2f aR1 = *(const v2f*)&sh_z1r[(16 + mrow) * ZSTR + ka];

        #pragma unroll
        for (int t = 0; t < 4; ++t) {
            const int n = n0 + t * 16 + ncol;
            v2f b;
            b.x = W2[(ka + 0) * DMID + n];
            b.y = W2[(ka + 1) * DMID + n];
            accF[0][t] = wmma_f32(aF0, b, accF[0][t]);
            accF[1][t] = wmma_f32(aF1, b, accF[1][t]);
            accR[0][t] = wmma_f32(aR0, b, accR[0][t]);
            accR[1][t] = wmma_f32(aR1, b, accR[1][t]);
        }
    }

    // ---- bias + relu + Janossy pool (fwd + rev), write pooled to LDS ----
    #pragma unroll
    for (int mt = 0; mt < 2; ++mt)
        #pragma unroll
        for (int t = 0; t < 4; ++t) {
            const int n = n0 + t * 16 + ncol;
            const float bias = b2[n];
            #pragma unroll
            for (int i = 0; i < 8; ++i) {
                const int m = mt * 16 + i + 8 * khalf;
                float vf = accF[mt][t][i] + bias; vf = vf > 0.f ? vf : 0.f;
                float vr = accR[mt][t][i] + bias; vr = vr > 0.f ? vr : 0.f;
                sh_p[m * ZSTR + n] = vf + vr;
            }
        }
    __syncthreads();

    // ---- stage 4: tiny output head pooled[32x256] @ Wout[256x3] + bout ----
    for (int o = tid; o < MBLOCK * 3; o += 128) {
        const int m = o / 3;
        const int d = o - m * 3;
        float s = bout[d];
        const float* prow = &sh_p[m * ZSTR];
        #pragma unroll 4
        for (int k = 0; k < DMID; ++k)
            s += prow[k] * Wout[k * 3 + d];
        out[(size_t)(frag0 + m) * 3 + d] = s;
    }
}

extern "C" void kernel_launch(void* const* d_in, const int* in_sizes, int n_in,
                              void* d_out, int out_size, void* d_ws, size_t ws_size,
                              hipStream_t stream) {
    const float* x_atoms = (const float*)d_in[0];
    const int*   frag_idx = (const int*)d_in[1];
    const float* W1   = (const float*)d_in[2];
    const float* b1   = (const float*)d_in[3];
    const float* W2   = (const float*)d_in[4];
    const float* b2   = (const float*)d_in[5];
    const float* Wout = (const float*)d_in[6];
    const float* bout = (const float*)d_in[7];
    float* out = (float*)d_out;

    const int n_frag = in_sizes[1] / 3;          // 262144
    const int grid = n_frag / MBLOCK;            // 8192 workgroups
    janossy_readout_kernel<<<grid, 128, 0, stream>>>(
        x_atoms, frag_idx, W1, b1, W2, b2, Wout, bout, out);
}